// TripletLoss_44513041056151
// MI455X (gfx1250) — compile-verified
//
#include <hip/hip_runtime.h>
#include <hip/hip_bf16.h>
#include <math.h>

// ---------------------------------------------------------------------------
// TripletLoss on MI455X (gfx1250, wave32).
//   N=4096 points, D=512, T=65536 triplets.
//   loss = mean_t softplus( d(i,j) - d(i,k) ),  d = clamped squared distance.
//
// Design: skip the N^2 GEMM of the reference. Per wave32, batch 16 triplets:
//   A (16x512) = xi rows, B (512x16) = xj (resp. xk) columns, and accumulate
//   with V_WMMA_F32_16X16X4_F32 (f32 in, f32 acc — full reference precision).
//   Diagonal of the 16x16 result = the 16 needed dot products.
// x (8 MB) lives entirely in the 192 MB L2, so all row gathers are L2 hits.
// ---------------------------------------------------------------------------

typedef __attribute__((ext_vector_type(2))) float v2f;
typedef __attribute__((ext_vector_type(8))) float v8f;

#define NPTS   4096
#define DIM    512
#define NTRIP  65536
#define NWAVE  (NTRIP / 16)   // 4096 waves, 16 triplets each

// ---- Kernel 1: row squared norms sq[n] = sum_d x[n][d]^2 (one wave per row)
__global__ __launch_bounds__(256) void tl_rownorm(const float* __restrict__ x,
                                                  float* __restrict__ sq) {
    const int wave = (blockIdx.x * blockDim.x + threadIdx.x) >> 5;
    const int lane = threadIdx.x & 31;
    if (wave >= NPTS) return;
    const float* row = x + (size_t)wave * DIM;
    float s = 0.f;
    #pragma unroll 4
    for (int c = lane; c < DIM; c += 32) {
        float v = row[c];
        s = fmaf(v, v, s);
    }
    #pragma unroll
    for (int off = 16; off > 0; off >>= 1) s += __shfl_xor(s, off, 32);
    if (lane == 0) sq[wave] = s;
}

// ---- Kernel 2: batched dot products via f32 WMMA + per-wave partial loss
__global__ __launch_bounds__(256) void tl_wmma(const float* __restrict__ x,
                                               const int*   __restrict__ trip,
                                               const float* __restrict__ sq,
                                               float* __restrict__ partials) {
    const int wave = (blockIdx.x * blockDim.x + threadIdx.x) >> 5;
    const int lane = threadIdx.x & 31;
    if (wave >= NWAVE) return;

    const int tbase = wave * 16;
    const int n     = lane & 15;          // matrix row (A) / column (B) index
    const int hi    = lane >> 4;          // 0: K+{0,1}, 1: K+{2,3}
    const int t     = tbase + n;

    // This lane's triplet indices (lanes 0-15 and 16-31 mirror each other).
    const int ii = trip[3 * t + 0];
    const int jj = trip[3 * t + 1];
    const int kk = trip[3 * t + 2];

    const float* __restrict__ rowA = x + (size_t)ii * DIM;  // A: xi rows
    const float* __restrict__ rowJ = x + (size_t)jj * DIM;  // B(pos): xj cols
    const float* __restrict__ rowK = x + (size_t)kk * DIM;  // B(neg): xk cols
    const int koff = hi * 2;

    v8f cP = {};  // accumulates 16x16 of xi . xj
    v8f cN = {};  // accumulates 16x16 of xi . xk

    // K loop: 4 per WMMA, unrolled x2 for load-level parallelism.
    #pragma unroll 2
    for (int k0 = 0; k0 < DIM; k0 += 4) {
        const v2f a  = *(const v2f*)(rowA + k0 + koff);
        const v2f bj = *(const v2f*)(rowJ + k0 + koff);
        const v2f bk = *(const v2f*)(rowK + k0 + koff);
        // 8 args: (neg_a, A, neg_b, B, c_mod, C, reuse_a, reuse_b)
        cP = __builtin_amdgcn_wmma_f32_16x16x4_f32(false, a, false, bj,
                                                   (short)0, cP, false, false);
        cN = __builtin_amdgcn_wmma_f32_16x16x4_f32(false, a, false, bk,
                                                   (short)0, cN, false, false);
    }

    // Diagonal extraction (M == N). f32 16x16 C/D layout:
    //   lane L, VGPR v  holds  M = v + 8*(L>=16), N = L%16.
    //   => diag t=0..7  at lanes 0..7,  v = lane
    //      diag t=8..15 at lanes 24..31, v = lane-24
    int sel = -1;
    if (lane < 8)        sel = lane;
    else if (lane >= 24) sel = lane - 24;

    float contrib = 0.f;
    if (sel >= 0) {
        float dp = 0.f, dn = 0.f;
        #pragma unroll
        for (int v = 0; v < 8; ++v) {
            if (sel == v) { dp = cP[v]; dn = cN[v]; }
        }
        const float si = sq[ii], sj = sq[jj], sk = sq[kk];
        const float dpos = fmaxf(si + sj - 2.f * dp, 0.f);
        const float dneg = fmaxf(si + sk - 2.f * dn, 0.f);
        const float z = dpos - dneg;
        // Numerically stable softplus: log1p(exp(z)) = max(z,0)+log1p(exp(-|z|))
        contrib = fmaxf(z, 0.f) + log1pf(__expf(-fabsf(z)));
    }

    // Deterministic wave reduction of the 16 lane contributions.
    #pragma unroll
    for (int off = 16; off > 0; off >>= 1) contrib += __shfl_xor(contrib, off, 32);
    if (lane == 0) partials[wave] = contrib;
}

// ---- Kernel 3: deterministic single-block reduction of 4096 partials
__global__ __launch_bounds__(256) void tl_reduce(const float* __restrict__ partials,
                                                 float* __restrict__ out) {
    __shared__ float smem[256];
    float v = 0.f;
    for (int i = threadIdx.x; i < NWAVE; i += 256) v += partials[i];
    smem[threadIdx.x] = v;
    __syncthreads();
    #pragma unroll
    for (int st = 128; st > 0; st >>= 1) {
        if (threadIdx.x < st) smem[threadIdx.x] += smem[threadIdx.x + st];
        __syncthreads();
    }
    if (threadIdx.x == 0) out[0] = smem[0] * (1.0f / (float)NTRIP);
}

extern "C" void kernel_launch(void* const* d_in, const int* in_sizes, int n_in,
                              void* d_out, int out_size, void* d_ws, size_t ws_size,
                              hipStream_t stream) {
    const float* x    = (const float*)d_in[0];   // (4096, 512) f32
    const int*   trip = (const int*)d_in[1];     // (65536, 3)  int
    float*       out  = (float*)d_out;           // scalar f32

    float* sq       = (float*)d_ws;              // 4096 floats
    float* partials = sq + NPTS;                 // 4096 floats

    // 1) row norms: 4096 waves, 8 waves/block -> 512 blocks
    tl_rownorm<<<NPTS / 8, 256, 0, stream>>>(x, sq);
    // 2) WMMA batched dots + per-wave partial losses: 4096 waves
    tl_wmma<<<NWAVE / 8, 256, 0, stream>>>(x, trip, sq, partials);
    // 3) final deterministic reduction
    tl_reduce<<<1, 256, 0, stream>>>(partials, out);
}